// TensorDiversity1_4286377361748
// MI455X (gfx1250) — compile-verified
//
#include <hip/hip_runtime.h>
#include <stdint.h>

// Problem constants (from the reference): B=16384 rows, L=1024 tokens/row,
// vocab V=50257, exclude tokens {0,1,2}. Output = mean over rows of
// (unique-count / L) as a single float32.
#define LROW    1024
#define VOCAB   50257
#define WORDS   ((VOCAB + 31) / 32)   // 1571 dwords = 6284 B bitmap
#define WORDS4  (WORDS / 4)           // 392 uint4's (1568 dwords), 3-dword tail
#define BLOCK   256                   // 8 wave32s per row
#define NSLOTS  256                   // global partial-sum slots in d_ws

typedef unsigned int u32;

// Pointer/vector types for the gfx1250 async global->LDS builtin, whose
// parameters are (v4i AS1*, v4i AS3*, imm offset, imm cpol).
typedef int v4i __attribute__((vector_size(16)));
typedef __attribute__((address_space(1))) v4i* gv4i_p;
typedef __attribute__((address_space(3))) v4i* lv4i_p;

#if defined(__HIP_DEVICE_COMPILE__) && \
    __has_builtin(__builtin_amdgcn_global_load_async_to_lds_b128)
#define HAVE_ASYNC_LDS 1
#else
#define HAVE_ASYNC_LDS 0
#endif

__global__ void __launch_bounds__(BLOCK) row_unique_kernel(
    const int* __restrict__ preds, u32* __restrict__ partials)
{
    __shared__ u32 toks[LROW];      // 4 KB: staged row (int32 tokens)
    __shared__ u32 bitmap[WORDS];   // 6284 B: presence bitmap over vocab
    __shared__ u32 blocksum;

    const int tid = threadIdx.x;
    const int* rowp = preds + (size_t)blockIdx.x * LROW;

    // ---- Stage the whole row into LDS: one async b128 per lane = 4 KB. ----
#if HAVE_ASYNC_LDS
    {
        const int byteoff = tid * 16;
        __builtin_amdgcn_global_load_async_to_lds_b128(
            (gv4i_p)((const char*)rowp + byteoff),
            (lv4i_p)((char*)toks + byteoff),
            /*offset=*/0, /*cpol=*/0);
    }
#else
    ((uint4*)toks)[tid] = ((const uint4*)rowp)[tid];
#endif

    // ---- Overlap the in-flight copy with clearing the bitmap (b128 stores).
    {
        const uint4 z4 = make_uint4(0u, 0u, 0u, 0u);
        for (int i = tid; i < WORDS4; i += BLOCK) ((uint4*)bitmap)[i] = z4;
        if (tid < (WORDS - WORDS4 * 4)) bitmap[WORDS4 * 4 + tid] = 0u;
        if (tid == 0) blocksum = 0u;
    }

#if HAVE_ASYNC_LDS
#if __has_builtin(__builtin_amdgcn_s_wait_asynccnt)
    __builtin_amdgcn_s_wait_asynccnt(0);
#else
    asm volatile("s_wait_asynccnt 0" ::: "memory");
#endif
#endif
    __syncthreads();

    // ---- Insert 4 tokens/thread; count new bits via ds_or_rtn pre-op value.
    //      Tokens {0,1,2} may set bits but are excluded by the predicate.
    const uint4 t4 = ((const uint4*)toks)[tid];
    u32 cnt = 0;
    {
        u32 m, old;
        m = 1u << (t4.x & 31u); old = atomicOr(&bitmap[t4.x >> 5], m);
        cnt += (u32)((t4.x > 2u) & ((old & m) == 0u));
        m = 1u << (t4.y & 31u); old = atomicOr(&bitmap[t4.y >> 5], m);
        cnt += (u32)((t4.y > 2u) & ((old & m) == 0u));
        m = 1u << (t4.z & 31u); old = atomicOr(&bitmap[t4.z >> 5], m);
        cnt += (u32)((t4.z > 2u) & ((old & m) == 0u));
        m = 1u << (t4.w & 31u); old = atomicOr(&bitmap[t4.w >> 5], m);
        cnt += (u32)((t4.w > 2u) & ((old & m) == 0u));
    }

    // ---- wave32 butterfly reduction, then cross-wave via LDS atomic. ----
    #pragma unroll
    for (int off = 16; off > 0; off >>= 1)
        cnt += __shfl_xor(cnt, off, 32);
    if ((tid & 31) == 0) atomicAdd(&blocksum, cnt);
    __syncthreads();

    // Deterministic integer accumulation, spread over 256 slots so the
    // 16384 global atomics don't serialize on one address.
    if (tid == 0) atomicAdd(&partials[blockIdx.x & (NSLOTS - 1)], blocksum);
}

__global__ void __launch_bounds__(NSLOTS) zero_partials_kernel(u32* partials) {
    partials[threadIdx.x] = 0u;
}

__global__ void __launch_bounds__(NSLOTS) finalize_kernel(
    const u32* __restrict__ partials, float* __restrict__ out, int rows)
{
    const int tid = threadIdx.x;
    u32 v = partials[tid];
    #pragma unroll
    for (int off = 16; off > 0; off >>= 1)
        v += __shfl_xor(v, off, 32);

    __shared__ u32 wsum[NSLOTS / 32];
    if ((tid & 31) == 0) wsum[tid >> 5] = v;
    __syncthreads();

    if (tid == 0) {
        u32 tot = 0;
        #pragma unroll
        for (int i = 0; i < NSLOTS / 32; ++i) tot += wsum[i];
        out[0] = (float)((double)tot / ((double)rows * (double)LROW));
    }
}

extern "C" void kernel_launch(void* const* d_in, const int* in_sizes, int n_in,
                              void* d_out, int out_size, void* d_ws, size_t ws_size,
                              hipStream_t stream) {
    const int* preds = (const int*)d_in[0];   // integer input -> const int*
    float* out = (float*)d_out;               // single f32 scalar
    u32* partials = (u32*)d_ws;               // 256 * 4 B of scratch

    const int total = in_sizes[0];            // B * L
    const int rows  = total / LROW;           // B = 16384

    zero_partials_kernel<<<1, NSLOTS, 0, stream>>>(partials);
    row_unique_kernel<<<rows, BLOCK, 0, stream>>>(preds, partials);
    finalize_kernel<<<1, NSLOTS, 0, stream>>>(partials, out, rows);
}